// SelfAttention_14456859918584
// MI455X (gfx1250) — compile-verified
//
#include <hip/hip_runtime.h>
#include <math.h>

// MI455X / gfx1250, wave32. GEMMs on the fp32 matrix pipe
// (V_WMMA_F32_16X16X4_F32) to match reference fp32 numerics. Tile staging uses
// CDNA5 async global->LDS DMA (GLOBAL_LOAD_ASYNC_TO_LDS_B128, ASYNCcnt) with
// double-buffered LDS so DMA overlaps WMMA.

typedef float v2f __attribute__((ext_vector_type(2)));
typedef float v8f __attribute__((ext_vector_type(8)));

#define BSZ   2
#define SEQ   2048
#define IND   1024
#define DHEAD 64
#define NHEAD 16

// Async 16B global->LDS copy. VDST = LDS byte-address VGPR (low 32 bits of a
// generic LDS pointer ARE the LDS offset per the aperture rules), VADDR = 64b
// global address, off = null SADDR. Tracked by ASYNCcnt.
#define ASYNC16(g, l)                                                      \
    asm volatile("global_load_async_to_lds_b128 %0, %1, off"               \
                 :: "v"((unsigned)(uintptr_t)(l)), "v"(g) : "memory")

#if __has_builtin(__builtin_amdgcn_s_wait_asynccnt)
#define WAIT_ASYNC(n) __builtin_amdgcn_s_wait_asynccnt((unsigned short)(n))
#else
#define WAIT_ASYNC(n) asm volatile("s_wait_asynccnt %0" :: "i"(n) : "memory")
#endif

__device__ __forceinline__ float sigm(float x) { return 1.0f / (1.0f + __expf(-x)); }

// ---------------------------------------------------------------------------
// Tiled fp32 WMMA GEMM:  out[4096 x N] = A[4096 x 1024] @ W[1024 x N] (+epilogue)
// Block: 128 threads (4 waves), 64x64 tile; async-DMA double-buffered staging.
// mode 0: out = gemm + bias
// mode 1: out = sigmoid(gemm + bias)
// mode 2: out = gemm + bias + resid
// ---------------------------------------------------------------------------
__global__ __launch_bounds__(128)
void gemm_wmma_f32(const float* __restrict__ A, const float* __restrict__ W,
                   const float* __restrict__ bias, const float* __restrict__ resid,
                   float* __restrict__ out, int N, int mode)
{
    __shared__ float As[2 * 64 * 20];   // pitch 20: 16B-aligned rows, conflict-free
    __shared__ float Bs[2 * 16 * 72];   // pitch 72: 16B-aligned rows, conflict-free

    const int tid  = threadIdx.x;
    const int w    = tid >> 5;
    const int lane = tid & 31;
    const int half = lane >> 4;
    const int l16  = lane & 15;
    const int m0   = blockIdx.y * 64;
    const int n0   = blockIdx.x * 64;

    auto stage = [&](int kt, int buf) {
        const int r  = tid >> 1;
        const int cp = (tid & 1) * 2;
        const float* ga = A + (size_t)(m0 + r) * 1024 + kt * 16 + cp * 4;
        float* la = &As[buf * 1280 + r * 20 + cp * 4];
        ASYNC16(ga,     la);
        ASYNC16(ga + 4, la + 4);
        const int rb = tid >> 3;
        const int cb = (tid & 7) * 2;
        const float* gb = W + (size_t)(kt * 16 + rb) * N + n0 + cb * 4;
        float* lb = &Bs[buf * 1152 + rb * 72 + cb * 4];
        ASYNC16(gb,     lb);
        ASYNC16(gb + 4, lb + 4);
    };

    v8f acc[4] = {};
    stage(0, 0);

    for (int kt = 0; kt < 64; ++kt) {   // K = 1024, 16 per step
        const int cur = kt & 1;
        WAIT_ASYNC(0);                  // my async writes for tile kt landed
        __syncthreads();                // everyone's landed; prev reads done
        if (kt + 1 < 64) stage(kt + 1, cur ^ 1);   // DMA overlaps WMMA below

        const float* Ac = &As[cur * 1280];
        const float* Bc = &Bs[cur * 1152];
#pragma unroll
        for (int kk = 0; kk < 4; ++kk) {
            const int ak = kk * 4 + 2 * half;
            v2f a;
            a.x = Ac[(w * 16 + l16) * 20 + ak];
            a.y = Ac[(w * 16 + l16) * 20 + ak + 1];
#pragma unroll
            for (int nt = 0; nt < 4; ++nt) {
                v2f b;
                b.x = Bc[ak * 72 + nt * 16 + l16];
                b.y = Bc[(ak + 1) * 72 + nt * 16 + l16];
                acc[nt] = __builtin_amdgcn_wmma_f32_16x16x4_f32(
                    false, a, false, b, (short)0, acc[nt], false, false);
            }
        }
    }

#pragma unroll
    for (int nt = 0; nt < 4; ++nt) {
#pragma unroll
        for (int j = 0; j < 8; ++j) {
            const int row = m0 + w * 16 + j + 8 * half;
            const int col = n0 + nt * 16 + l16;
            float v = acc[nt][j] + bias[col];
            if (mode == 1)      v = sigm(v);
            else if (mode == 2) v += resid[(size_t)row * N + col];
            out[(size_t)row * N + col] = v;
        }
    }
}

// ---------------------------------------------------------------------------
// Bout[4096 x 16] = x @ Wb + bb   (tiny; scalar)
// ---------------------------------------------------------------------------
__global__ __launch_bounds__(64)
void bgate(const float* __restrict__ x, const float* __restrict__ Wb,
           const float* __restrict__ bb, float* __restrict__ Bout)
{
    const int r = blockIdx.x * 4 + (threadIdx.x >> 4);
    const int h = threadIdx.x & 15;
    float acc = bb[h];
    const float* xr = x + (size_t)r * IND;
    for (int k = 0; k < IND; ++k) acc += xr[k] * Wb[k * NHEAD + h];
    Bout[(size_t)r * NHEAD + h] = acc;
}

// ---------------------------------------------------------------------------
// Split heads (channel = d*16 + h), fold sigmoid(alpha) scales, apply the
// reference rotation (odd uses the NEW even), relayout to [b,h,s,64].
// ---------------------------------------------------------------------------
__global__ __launch_bounds__(512)
void rope_split(const float* __restrict__ QKV, const float* __restrict__ rcos,
                const float* __restrict__ rsin, const float* __restrict__ alphaq,
                const float* __restrict__ alphak, const float* __restrict__ alphav,
                float* __restrict__ Qr, float* __restrict__ Kr, float* __restrict__ Vr)
{
    const int tid = threadIdx.x;
    const int h = tid & 15, i = tid >> 4;
    const int row = blockIdx.x;           // b*SEQ + s
    const int b = row >> 11, s = row & (SEQ - 1);

    const float aq = sigm(alphaq[0]);
    const float ak = sigm(alphak[0]);
    const float av = sigm(alphav[0]);
    const float c0 = rcos[s * DHEAD + 2 * i];
    const float c1 = rcos[s * DHEAD + 2 * i + 1];
    const float sn = rsin[s * DHEAD + 2 * i];   // rsin[:,0::2][i]

    const size_t base = (size_t)row * 3072;
    const int ce = 32 * i + h;            // channel of d=2i
    const int co = 32 * i + 16 + h;       // channel of d=2i+1
    const size_t ob = ((size_t)(b * NHEAD + h) * SEQ + s) * DHEAD;

    {
        float xe = QKV[base + ce] * aq * c0;
        float xo = QKV[base + co] * aq * c1;
        float e = xe - sn * xo;
        float o = xo + sn * e;
        Qr[ob + 2 * i] = e; Qr[ob + 2 * i + 1] = o;
    }
    {
        float xe = QKV[base + IND + ce] * ak * c0;
        float xo = QKV[base + IND + co] * ak * c1;
        float e = xe - sn * xo;
        float o = xo + sn * e;
        Kr[ob + 2 * i] = e; Kr[ob + 2 * i + 1] = o;
    }
    Vr[ob + 2 * i]     = QKV[base + 2 * IND + ce] * av;
    Vr[ob + 2 * i + 1] = QKV[base + 2 * IND + co] * av;
}

// ---------------------------------------------------------------------------
// Flash attention per (b,h). 4 waves x 16 query rows; async-DMA double-buffered
// K/V tiles; S and P.V on the fp32 WMMA pipe; online softmax via width-16
// shfl_xor; P reshaped C-layout -> A-layout through a per-wave LDS tile.
// Output gated by G, written over the G buffer in [b, s, d*16+h] layout.
// ---------------------------------------------------------------------------
__global__ __launch_bounds__(128)
void flash_attn(const float* __restrict__ Qr, const float* __restrict__ Kr,
                const float* __restrict__ Vr, const float* __restrict__ Bb,
                const float* __restrict__ mask, float* __restrict__ Gatt)
{
    __shared__ float Ks[2 * 16 * 68];     // [key][d] pitch 68 (16B-aligned rows)
    __shared__ float Vs[2 * 16 * 72];     // [key][d] pitch 72 (16B-aligned rows)
    __shared__ float Pl[4 * 16 * 18];     // per-wave P tile, pitch 18
    __shared__ float bbs[2 * 16];

    const int tid  = threadIdx.x;
    const int w    = tid >> 5;
    const int lane = tid & 31;
    const int half = lane >> 4;
    const int l16  = lane & 15;
    const int bh = blockIdx.y;
    const int b  = bh >> 4, h = bh & 15;
    const int q0 = blockIdx.x * 64 + w * 16;

    const float* Qb = Qr + (size_t)bh * SEQ * DHEAD;
    const float* Kb = Kr + (size_t)bh * SEQ * DHEAD;
    const float* Vb = Vr + (size_t)bh * SEQ * DHEAD;

    auto stage = [&](int kt, int buf) {
        const int key = tid >> 3, cp = tid & 7;
        ASYNC16(Kb + (size_t)(kt * 16 + key) * DHEAD + cp * 4,
                &Ks[buf * 1088 + key * 68 + cp * 4]);
        ASYNC16(Vb + (size_t)(kt * 16 + key) * DHEAD + cp * 4,
                &Vs[buf * 1152 + key * 72 + cp * 4]);
        if (tid < 16) {
            const int kg = kt * 16 + tid;
            bbs[buf * 16 + tid] =
                Bb[(size_t)(b * SEQ + kg) * NHEAD + h] - mask[b * SEQ + kg];
        }
    };

    // Resident Q fragments: 16 k-steps of 4 over d=0..63
    v2f qf[16];
#pragma unroll
    for (int kk = 0; kk < 16; ++kk) {
        const float* p = Qb + (size_t)(q0 + l16) * DHEAD + kk * 4 + 2 * half;
        qf[kk].x = p[0]; qf[kk].y = p[1];
    }

    v8f acc[4] = {};
    float mrow[8], lrow[8];
#pragma unroll
    for (int j = 0; j < 8; ++j) { mrow[j] = -1e30f; lrow[j] = 0.0f; }

    const int pbase = w * 16 * 18;
    stage(0, 0);

    for (int kt = 0; kt < SEQ / 16; ++kt) {
        const int cur = kt & 1;
        WAIT_ASYNC(0);                 // tile kt in LDS (mine)
        __syncthreads();               // all waves' tile kt in; prev reads done
        if (kt + 1 < SEQ / 16) stage(kt + 1, cur ^ 1);   // DMA overlaps compute

        const float* Kc = &Ks[cur * 1088];
        const float* Vc = &Vs[cur * 1152];

        // S tile (16q x 16k) = Q . K^T
        v8f s = {};
#pragma unroll
        for (int kk = 0; kk < 16; ++kk) {
            const int ak = kk * 4 + 2 * half;
            v2f bf;
            bf.x = Kc[l16 * 68 + ak];
            bf.y = Kc[l16 * 68 + ak + 1];
            s = __builtin_amdgcn_wmma_f32_16x16x4_f32(
                false, qf[kk], false, bf, (short)0, s, false, false);
        }

        // Online softmax (rows j+8*half live on one lane-half)
        const float bc = bbs[cur * 16 + l16];
#pragma unroll
        for (int j = 0; j < 8; ++j) {
            float v = s[j] * 0.125f + bc;           // scale = D^-0.5
            float r = v;
            r = fmaxf(r, __shfl_xor(r, 1, 16));
            r = fmaxf(r, __shfl_xor(r, 2, 16));
            r = fmaxf(r, __shfl_xor(r, 4, 16));
            r = fmaxf(r, __shfl_xor(r, 8, 16));
            const float mn = fmaxf(mrow[j], r);
            const float al = __expf(mrow[j] - mn);
            const float p  = __expf(v - mn);
            float su = p;
            su += __shfl_xor(su, 1, 16);
            su += __shfl_xor(su, 2, 16);
            su += __shfl_xor(su, 4, 16);
            su += __shfl_xor(su, 8, 16);
            lrow[j] = lrow[j] * al + su;
            mrow[j] = mn;
#pragma unroll
            for (int nt = 0; nt < 4; ++nt) acc[nt][j] *= al;
            Pl[pbase + (j + 8 * half) * 18 + l16] = p;   // C-layout -> LDS
        }
        __syncthreads();   // Pl visible cross-lane; Vs[cur] stable

        // O += P . V
#pragma unroll
        for (int kk = 0; kk < 4; ++kk) {
            const int ak = kk * 4 + 2 * half;
            v2f a;
            a.x = Pl[pbase + l16 * 18 + ak];
            a.y = Pl[pbase + l16 * 18 + ak + 1];
#pragma unroll
            for (int nt = 0; nt < 4; ++nt) {
                v2f bf;
                bf.x = Vc[ak * 72 + nt * 16 + l16];
                bf.y = Vc[(ak + 1) * 72 + nt * 16 + l16];
                acc[nt] = __builtin_amdgcn_wmma_f32_16x16x4_f32(
                    false, a, false, bf, (short)0, acc[nt], false, false);
            }
        }
    }

    // Normalize, gate by G, write in [b, s, d*16+h] layout (over G buffer)
#pragma unroll
    for (int j = 0; j < 8; ++j) {
        const float rn = 1.0f / lrow[j];
        const int q = q0 + j + 8 * half;
#pragma unroll
        for (int nt = 0; nt < 4; ++nt) {
            const int dd = nt * 16 + l16;
            const size_t idx = (size_t)(b * SEQ + q) * IND + dd * NHEAD + h;
            Gatt[idx] = acc[nt][j] * rn * Gatt[idx];
        }
    }
}

// ---------------------------------------------------------------------------
extern "C" void kernel_launch(void* const* d_in, const int* in_sizes, int n_in,
                              void* d_out, int out_size, void* d_ws, size_t ws_size,
                              hipStream_t stream)
{
    (void)in_sizes; (void)n_in; (void)out_size; (void)ws_size;
    const float* x      = (const float*)d_in[0];
    const float* mask   = (const float*)d_in[1];
    const float* Wqkv   = (const float*)d_in[2];
    const float* bqkv   = (const float*)d_in[3];
    const float* Wo     = (const float*)d_in[4];
    const float* bo     = (const float*)d_in[5];
    const float* Wg     = (const float*)d_in[6];
    const float* bg     = (const float*)d_in[7];
    const float* Wb     = (const float*)d_in[8];
    const float* bb     = (const float*)d_in[9];
    const float* alphaq = (const float*)d_in[10];
    const float* alphak = (const float*)d_in[11];
    const float* alphav = (const float*)d_in[12];
    const float* rcos   = (const float*)d_in[13];
    const float* rsin   = (const float*)d_in[14];
    float* out = (float*)d_out;

    // Workspace layout (floats)
    float* ws   = (float*)d_ws;
    float* QKV  = ws;                                   // 4096*3072
    float* G    = QKV  + (size_t)4096 * 3072;           // 4096*1024 (becomes ATT)
    float* Bout = G    + (size_t)4096 * 1024;           // 4096*16
    float* Qr   = Bout + (size_t)4096 * 16;             // 32*2048*64
    float* Kr   = Qr   + (size_t)32 * SEQ * DHEAD;
    float* Vr   = Kr   + (size_t)32 * SEQ * DHEAD;

    const dim3 blk(128);
    gemm_wmma_f32<<<dim3(3072 / 64, 4096 / 64), blk, 0, stream>>>(
        x, Wqkv, bqkv, nullptr, QKV, 3072, 0);
    gemm_wmma_f32<<<dim3(1024 / 64, 4096 / 64), blk, 0, stream>>>(
        x, Wg, bg, nullptr, G, 1024, 1);
    bgate<<<1024, 64, 0, stream>>>(x, Wb, bb, Bout);
    rope_split<<<BSZ * SEQ, 512, 0, stream>>>(
        QKV, rcos, rsin, alphaq, alphak, alphav, Qr, Kr, Vr);
    flash_attn<<<dim3(SEQ / 64, BSZ * NHEAD), blk, 0, stream>>>(
        Qr, Kr, Vr, Bout, mask, G);
    gemm_wmma_f32<<<dim3(1024 / 64, 4096 / 64), blk, 0, stream>>>(
        G, Wo, bo, x, out, 1024, 2);
}